// CoAttentionLayer_6914897347236
// MI455X (gfx1250) — compile-verified
//
#include <hip/hip_runtime.h>

// ---------------------------------------------------------------------------
// CoAttention (no softmax) for MI455X / gfx1250, fp32 WMMA (V_WMMA_F32_16X16X4_F32)
//
// Shapes (fixed by reference setup): B=4, L=2048, C=1024, H=16, D=64, coatten=1024
// Reassociation: o_half = q_half @ (scale * k_other^T @ v_other)   [64x64 per bh]
// ---------------------------------------------------------------------------

typedef float v2f __attribute__((ext_vector_type(2)));
typedef float v8f __attribute__((ext_vector_type(8)));

#define WMMA_F32(a, b, c) \
  __builtin_amdgcn_wmma_f32_16x16x4_f32(false, (a), false, (b), (short)0, (c), false, false)

// Workspace layout (in floats)
#define Q_OFF   0u
#define K_OFF   8388608u      // 8192*1024
#define V_OFF   16777216u
#define KV_OFF  25165824u     // + B*H*2*64*64 = 524288
#define O_OFF   25690112u     // + 8192*1024

// ---------------------------------------------------------------------------
// Block-tiled 256x128 fp32 WMMA GEMM, K=1024, row-major A[MxK] * B[KxN].
// Double-buffered LDS: next tile's global loads overlap current tile's WMMAs;
// one barrier per K-step.
// 8 waves: wave = (waveM 0..3) x (waveN 0..1); each wave computes 64x64
// via 4x4 accumulators of 16x16 -> 16 WMMA per 8 fragment loads.
// EPI==0: scatter into q/k/v [part][b][h][l][d]   (N must be 3072)
// EPI==1: plain row-major out[M x N] + bias[c]
// ---------------------------------------------------------------------------
template <int EPI>
__global__ __launch_bounds__(256) void gemm256(const float* __restrict__ A,
                                               const float* __restrict__ Bm,
                                               float* __restrict__ out,
                                               const float* __restrict__ bias,
                                               int N) {
  __shared__ __align__(16) float As[2][256][17];   // [m][k], stride 17 -> conflict-free frag reads
  __shared__ __align__(16) float Bs[2][16][136];   // [k][n], stride 136 -> half-waves on disjoint banks

  const int t = threadIdx.x;
  const int lane = t & 31;
  const int wave = t >> 5;
  const int rowBase = blockIdx.x * 256;
  const int colBase = blockIdx.y * 128;
  const int K = 1024;
  const int waveM = (wave >> 1) * 64;
  const int waveN = (wave & 1) * 64;
  const int l15 = lane & 15;
  const int koff = (lane >> 4) * 2;

  // Loader geometry (per thread): A tile 256x16 = 1024 float4 (4 each),
  // B tile 16x128 = 512 float4 (2 each).
  const int arow = t >> 2;          // 0..63 (+ it*64)
  const int ac4 = (t & 3) * 4;      // 0,4,8,12
  const int brow = t >> 5;          // 0..7 (+ it*8)
  const int bcol = (t & 31) * 4;    // 0..124

  float4 ga[4], gb[2];

  auto loadG = [&](int k0) {
#pragma unroll
    for (int it = 0; it < 4; ++it)
      ga[it] = *reinterpret_cast<const float4*>(
          &A[(size_t)(rowBase + arow + it * 64) * K + k0 + ac4]);
#pragma unroll
    for (int it = 0; it < 2; ++it)
      gb[it] = *reinterpret_cast<const float4*>(
          &Bm[(size_t)(k0 + brow + it * 8) * N + colBase + bcol]);
  };
  auto storeL = [&](int buf) {
#pragma unroll
    for (int it = 0; it < 4; ++it) {
      As[buf][arow + it * 64][ac4 + 0] = ga[it].x;
      As[buf][arow + it * 64][ac4 + 1] = ga[it].y;
      As[buf][arow + it * 64][ac4 + 2] = ga[it].z;
      As[buf][arow + it * 64][ac4 + 3] = ga[it].w;
    }
#pragma unroll
    for (int it = 0; it < 2; ++it)
      *reinterpret_cast<float4*>(&Bs[buf][brow + it * 8][bcol]) = gb[it];
  };

  v8f acc[4][4] = {};

  loadG(0);
  storeL(0);
  __syncthreads();

  for (int k0 = 0; k0 < K; k0 += 16) {
    const int cur = (k0 >> 4) & 1;
    const bool more = (k0 + 16) < K;
    if (more) loadG(k0 + 16);   // in flight during the 64 WMMAs below

#pragma unroll
    for (int kk = 0; kk < 16; kk += 4) {
      v2f af[4];
#pragma unroll
      for (int mt = 0; mt < 4; ++mt) {
        int m = waveM + mt * 16 + l15;
        af[mt].x = As[cur][m][kk + koff];
        af[mt].y = As[cur][m][kk + koff + 1];
      }
      v2f bf[4];
#pragma unroll
      for (int nt = 0; nt < 4; ++nt) {
        int n = waveN + nt * 16 + l15;
        bf[nt].x = Bs[cur][kk + koff][n];
        bf[nt].y = Bs[cur][kk + koff + 1][n];
      }
#pragma unroll
      for (int mt = 0; mt < 4; ++mt)
#pragma unroll
        for (int nt = 0; nt < 4; ++nt)
          acc[mt][nt] = WMMA_F32(af[mt], bf[nt], acc[mt][nt]);
    }

    if (more) storeL(cur ^ 1);
    __syncthreads();   // single barrier per K-step
  }

  // Epilogue. C/D layout: VGPR j holds (M=j, lanes 0-15) and (M=j+8, lanes 16-31), N=lane&15.
#pragma unroll
  for (int mt = 0; mt < 4; ++mt) {
#pragma unroll
    for (int nt = 0; nt < 4; ++nt) {
      int c = colBase + waveN + nt * 16 + l15;
      int rbase = rowBase + waveM + mt * 16 + ((lane >> 4) * 8);
      if (EPI == 0) {
        int part = c >> 10;          // 0=q 1=k 2=v
        int cc = c & 1023;
        int h = cc >> 6;
        int d = cc & 63;
#pragma unroll
        for (int j = 0; j < 8; ++j) {
          int r = rbase + j;
          int b = r >> 11;           // L = 2048
          int l = r & 2047;
          size_t o = (size_t)part * 8388608u +
                     (((size_t)(b * 16 + h) * 2048 + l) * 64) + d;
          out[o] = acc[mt][nt][j];
        }
      } else {
        float bv = bias[c];
#pragma unroll
        for (int j = 0; j < 8; ++j) {
          int r = rbase + j;
          out[(size_t)r * N + c] = acc[mt][nt][j] + bv;
        }
      }
    }
  }
}

// ---------------------------------------------------------------------------
// KV[b][h][half] = scale * k_other^T @ v_other   (64x64, K=1024)
// One block per (bh, half). 8 waves: wave -> (mtile 0..3) x (pair of ntiles).
// ---------------------------------------------------------------------------
__global__ __launch_bounds__(256) void kv_kernel(const float* __restrict__ kmat,
                                                 const float* __restrict__ vmat,
                                                 float* __restrict__ kv) {
  __shared__ __align__(16) float As[64][17];   // As[d1][k] = k[l0+k][d1] (transposed)
  __shared__ __align__(16) float Bs[16][72];   // Bs[k][d2] = v[l0+k][d2]

  const int t = threadIdx.x;
  const int lane = t & 31;
  const int wave = t >> 5;
  const int id = blockIdx.x;        // 0..127
  const int half = id & 1;
  const int bh = id >> 1;
  const int srcHalf = 1 - half;     // q of `half` attends k/v of other half
  const float* kp = kmat + (size_t)bh * 131072u + (size_t)srcHalf * 65536u;
  const float* vp = vmat + (size_t)bh * 131072u + (size_t)srcHalf * 65536u;
  const int l15 = lane & 15;
  const int koff = (lane >> 4) * 2;
  const int m0 = (wave >> 1) * 16;
  const int n0 = (wave & 1) * 32;

  v8f acc[2] = {};

  for (int k0 = 0; k0 < 1024; k0 += 16) {
    int row = t >> 4;          // 0..15  (k within step)
    int c4 = (t & 15) * 4;     // 0..60  (d)
    float4 gk = *reinterpret_cast<const float4*>(&kp[(size_t)(k0 + row) * 64 + c4]);
    As[c4 + 0][row] = gk.x;
    As[c4 + 1][row] = gk.y;
    As[c4 + 2][row] = gk.z;
    As[c4 + 3][row] = gk.w;
    float4 gv = *reinterpret_cast<const float4*>(&vp[(size_t)(k0 + row) * 64 + c4]);
    *reinterpret_cast<float4*>(&Bs[row][c4]) = gv;
    __syncthreads();

#pragma unroll
    for (int kk = 0; kk < 16; kk += 4) {
      v2f af;
      int m = m0 + l15;
      af.x = As[m][kk + koff];
      af.y = As[m][kk + koff + 1];
#pragma unroll
      for (int nt = 0; nt < 2; ++nt) {
        v2f bf;
        int n = n0 + nt * 16 + l15;
        bf.x = Bs[kk + koff][n];
        bf.y = Bs[kk + koff + 1][n];
        acc[nt] = WMMA_F32(af, bf, acc[nt]);
      }
    }
    __syncthreads();
  }

  const float scale = 0.125f;  // D=64 -> D^-0.5
  float* dst = kv + ((size_t)bh * 2 + half) * 4096u;
#pragma unroll
  for (int nt = 0; nt < 2; ++nt) {
#pragma unroll
    for (int j = 0; j < 8; ++j) {
      int m = m0 + j + ((lane >> 4) * 8);
      int n = n0 + nt * 16 + l15;
      dst[m * 64 + n] = acc[nt][j] * scale;
    }
  }
}

// ---------------------------------------------------------------------------
// o rows: o[b][l][h*64+n] = q[b][h][l][:] @ KV[b][h][half]
// One block per (bh, half, 128-row mtile). KV kept fully in LDS.
// Wave w handles 16 rows x 64 cols (4 accumulators).
// ---------------------------------------------------------------------------
__global__ __launch_bounds__(256) void attn_o_kernel(const float* __restrict__ q,
                                                     const float* __restrict__ kv,
                                                     float* __restrict__ o) {
  __shared__ __align__(16) float Bs[64][72];

  const int t = threadIdx.x;
  const int lane = t & 31;
  const int wave = t >> 5;
  int id = blockIdx.x;              // 0..1023
  const int mt = id & 7;            // 128-row tile within the 1024-row half
  id >>= 3;
  const int half = id & 1;
  const int bh = id >> 1;
  const int b = bh >> 4;
  const int h = bh & 15;
  const int l15 = lane & 15;
  const int koff = (lane >> 4) * 2;

  const float* kvp = kv + ((size_t)bh * 2 + half) * 4096u;
#pragma unroll
  for (int i = 0; i < 4; ++i) {
    int idx = t + i * 256;          // 0..1023 float4s over 64x64
    int row = idx >> 4;
    int c4 = (idx & 15) * 4;
    float4 g = *reinterpret_cast<const float4*>(&kvp[row * 64 + c4]);
    *reinterpret_cast<float4*>(&Bs[row][c4]) = g;
  }
  __syncthreads();

  const int l0 = half * 1024 + mt * 128;
  const float* qp = q + (size_t)bh * 131072u + (size_t)(l0 + wave * 16 + l15) * 64;

  v8f acc[4] = {};
#pragma unroll
  for (int kk = 0; kk < 64; kk += 4) {
    v2f af;
    af.x = qp[kk + koff];
    af.y = qp[kk + koff + 1];
#pragma unroll
    for (int nt = 0; nt < 4; ++nt) {
      v2f bf;
      int n = nt * 16 + l15;
      bf.x = Bs[kk + koff][n];
      bf.y = Bs[kk + koff + 1][n];
      acc[nt] = WMMA_F32(af, bf, acc[nt]);
    }
  }

#pragma unroll
  for (int nt = 0; nt < 4; ++nt) {
#pragma unroll
    for (int j = 0; j < 8; ++j) {
      int l = l0 + wave * 16 + j + ((lane >> 4) * 8);
      int n = nt * 16 + l15;
      o[((size_t)b * 2048 + l) * 1024 + h * 64 + n] = acc[nt][j];
    }
  }
}

// ---------------------------------------------------------------------------
extern "C" void kernel_launch(void* const* d_in, const int* in_sizes, int n_in,
                              void* d_out, int out_size, void* d_ws, size_t ws_size,
                              hipStream_t stream) {
  (void)in_sizes; (void)n_in; (void)out_size; (void)ws_size;
  const float* x     = (const float*)d_in[0];   // [4,2048,1024]
  const float* Wqkv  = (const float*)d_in[1];   // [1024,3072]
  const float* Wproj = (const float*)d_in[2];   // [1024,1024]
  const float* bproj = (const float*)d_in[3];   // [1024]
  // d_in[4] = coatten (=1024, baked into the kernels)

  float* ws = (float*)d_ws;
  float* qkv = ws;                 // q at +0, k at +K_OFF, v at +V_OFF
  float* kp  = ws + K_OFF;
  float* vp  = ws + V_OFF;
  float* kvp = ws + KV_OFF;
  float* op  = ws + O_OFF;
  float* out = (float*)d_out;

  // 1) QKV GEMM (8192x3072, K=1024), scatter into [b][h][l][d] per q/k/v
  gemm256<0><<<dim3(32, 24), 256, 0, stream>>>(x, Wqkv, qkv, nullptr, 3072);
  // 2) KV = scale * k_other^T @ v_other  (64x64 per bh-half)
  kv_kernel<<<128, 256, 0, stream>>>(kp, vp, kvp);
  // 3) o = q_half @ KV
  attn_o_kernel<<<1024, 256, 0, stream>>>(ws /*q*/, kvp, op);
  // 4) out = o @ W_proj + b_proj  (8192x1024, K=1024)
  gemm256<1><<<dim3(32, 8), 256, 0, stream>>>(op, Wproj, out, bproj, 1024);
}